// MyModel_78039555769135
// MI455X (gfx1250) — compile-verified
//
#include <hip/hip_runtime.h>

typedef __attribute__((ext_vector_type(2))) float v2f;
typedef __attribute__((ext_vector_type(8))) float v8f;

namespace {
constexpr int B_ = 4;
constexpr int C_ = 3;
constexpr int H_ = 2048;
constexpr int W_ = 2048;
constexpr int CH_STRIDE = H_ * W_;             // img channel stride
constexpr int CO_STRIDE = (H_ - 1) * W_;       // == H_*(W_-1): obj channel stride (both branches)
constexpr int NTW = W_ / 16;                   // 128 16-pixel tiles per image row
constexpr int NTILES = B_ * H_ * NTW;          // 2^20 tiles
constexpr int THREADS = 256;                   // 8 wave32s
constexpr int BLOCKS = 1024;
}

__global__ __launch_bounds__(THREADS) void grad_loss_partial(
    const float* __restrict__ img,
    const float* __restrict__ objx,
    const float* __restrict__ objy,
    const float* __restrict__ kx,
    const float* __restrict__ ky,
    const float* __restrict__ mask,
    float* __restrict__ partial)
{
  const int lane = threadIdx.x & 31;
  const int wib  = threadIdx.x >> 5;
  const int n    = lane & 15;          // pixel / N column within tile
  const int half = lane >> 4;          // which half-wave
  const int waveId   = blockIdx.x * (THREADS / 32) + wib;
  const int numWaves = gridDim.x * (THREADS / 32);

  // ---- A matrices (weights) in 16x4 f32 WMMA layout:
  // lane m = lane&15; VGPR0 = A[m][2*half], VGPR1 = A[m][2*half+1].
  // K rows of WMMA#1: (ci,tap) = (0,0),(0,1),(1,0),(1,1)  -> kflat[m*6 + k]
  // K rows of WMMA#2: (2,0),(2,1),0,0                      -> kflat[m*6 + 4 + k]
  const int m = n;
  float ax0 = 0.f, ax1 = 0.f, ax20 = 0.f, ax21 = 0.f;
  float ay0 = 0.f, ay1 = 0.f, ay20 = 0.f, ay21 = 0.f;
  if (m < C_) {
    ax0 = kx[m * 6 + 2 * half + 0];
    ax1 = kx[m * 6 + 2 * half + 1];
    ay0 = ky[m * 6 + 2 * half + 0];
    ay1 = ky[m * 6 + 2 * half + 1];
    if (half == 0) {           // K=0,1 of WMMA#2; K=2,3 are zero padding
      ax20 = kx[m * 6 + 4]; ax21 = kx[m * 6 + 5];
      ay20 = ky[m * 6 + 4]; ay21 = ky[m * 6 + 5];
    }
  }
  const v2f aX1 = {ax0, ax1}, aX2 = {ax20, ax21};
  const v2f aY1 = {ay0, ay1}, aY2 = {ay20, ay21};

  float accX = 0.f, accY = 0.f;

  for (int t = waveId; t < NTILES; t += numWaves) {
    const int tw = t & (NTW - 1);
    const int h  = (t >> 7) & (H_ - 1);
    const int b  = t >> 18;
    const int w  = (tw << 4) + n;

    const int imgBase = b * C_ * CH_STRIDE;

    // B operand, 4x16 f32: VGPR0 holds rows {0,1} (lanes 0-15 / 16-31),
    // VGPR1 holds rows {2,3}. Row k=(ci,tap): this lane supplies column n of
    // row (ci = VGPR slot, tap = half).
    const int wt = w + half;                       // <= 2048: at w=W-1 reads one past the row (masked)
    const int rX = imgBase + h * W_ + wt;
    const v2f bX1 = { img[rX], img[rX + CH_STRIDE] };        // ci=0, ci=1 @ tap=half
    const v2f bX2 = { img[rX + 2 * CH_STRIDE], 0.f };        // ci=2 ; rows 2,3 zero

    const int hy = (h + half < H_) ? (h + half) : (H_ - 1);  // clamp; masked later
    const int rY = imgBase + hy * W_ + w;
    const v2f bY1 = { img[rY], img[rY + CH_STRIDE] };
    const v2f bY2 = { img[rY + 2 * CH_STRIDE], 0.f };

    // D[co][pixel] = sum_{ci,tap} K[co][ci,tap] * img_patch
    v8f dX = {};
    dX = __builtin_amdgcn_wmma_f32_16x16x4_f32(false, aX1, false, bX1, (short)0, dX, false, false);
    dX = __builtin_amdgcn_wmma_f32_16x16x4_f32(false, aX2, false, bX2, (short)0, dX, false, false);
    v8f dY = {};
    dY = __builtin_amdgcn_wmma_f32_16x16x4_f32(false, aY1, false, bY1, (short)0, dY, false, false);
    dY = __builtin_amdgcn_wmma_f32_16x16x4_f32(false, aY2, false, bY2, (short)0, dY, false, false);

    // D rows co=0..2 for pixel n live in VGPR co, lane n. Ship the y-branch
    // results to the upper half-wave so all 32 lanes do L1+reduce work.
    const float y0 = __shfl(dY[0], n);
    const float y1 = __shfl(dY[1], n);
    const float y2 = __shfl(dY[2], n);
    const float s0 = half ? y0 : dX[0];
    const float s1 = half ? y1 : dX[1];
    const float s2 = half ? y2 : dX[2];

    const float mv = mask[(b * H_ + h) * W_ + w];
    const int wc = (w < W_ - 1) ? w : (W_ - 2);    // clamped edge (masked to 0)
    const int hc = (h < H_ - 1) ? h : (H_ - 2);
    const int idxX = (b * C_ * H_ + h) * (W_ - 1) + wc;
    const int idxY = (b * C_ * (H_ - 1) + hc) * W_ + w;
    const float* objp = half ? objy : objx;
    const int idx0 = half ? idxY : idxX;
    const float o0 = objp[idx0];
    const float o1 = objp[idx0 + CO_STRIDE];
    const float o2 = objp[idx0 + 2 * CO_STRIDE];

    float term = fabsf((s0 - o0) * mv) + fabsf((s1 - o1) * mv) + fabsf((s2 - o2) * mv);
    const bool valid = half ? (h < H_ - 1) : (w < W_ - 1);
    term = valid ? term : 0.f;
    accX += half ? 0.f : term;
    accY += half ? term : 0.f;
  }

  // wave32 butterfly reduce
  for (int off = 16; off > 0; off >>= 1) {
    accX += __shfl_xor(accX, off);
    accY += __shfl_xor(accY, off);
  }
  __shared__ float sX[THREADS / 32], sY[THREADS / 32];
  if (lane == 0) { sX[wib] = accX; sY[wib] = accY; }
  __syncthreads();
  if (threadIdx.x == 0) {
    float tx = 0.f, ty = 0.f;
    for (int i = 0; i < THREADS / 32; ++i) { tx += sX[i]; ty += sY[i]; }
    partial[2 * blockIdx.x]     = tx;
    partial[2 * blockIdx.x + 1] = ty;
  }
}

__global__ __launch_bounds__(256) void grad_loss_finish(
    const float* __restrict__ partial, float* __restrict__ out, int nparts)
{
  __shared__ float sx[256], sy[256];
  float ax = 0.f, ay = 0.f;
  for (int i = threadIdx.x; i < nparts; i += 256) {
    ax += partial[2 * i];
    ay += partial[2 * i + 1];
  }
  sx[threadIdx.x] = ax;
  sy[threadIdx.x] = ay;
  __syncthreads();
  for (int s = 128; s > 0; s >>= 1) {
    if (threadIdx.x < s) {
      sx[threadIdx.x] += sx[threadIdx.x + s];
      sy[threadIdx.x] += sy[threadIdx.x + s];
    }
    __syncthreads();
  }
  if (threadIdx.x == 0) {
    // N = B*C*H*(W-1) = B*C*(H-1)*W = 50,307,072 for both means
    const float invN = 1.0f / 50307072.0f;
    out[0] = sx[0] * invN + sy[0] * invN;
  }
}

extern "C" void kernel_launch(void* const* d_in, const int* in_sizes, int n_in,
                              void* d_out, int out_size, void* d_ws, size_t ws_size,
                              hipStream_t stream) {
  const float* img  = (const float*)d_in[0];
  const float* objx = (const float*)d_in[1];
  const float* objy = (const float*)d_in[2];
  const float* kx   = (const float*)d_in[3];
  const float* ky   = (const float*)d_in[4];
  const float* mask = (const float*)d_in[5];
  float* out     = (float*)d_out;
  float* partial = (float*)d_ws;   // needs 2*BLOCKS floats = 8 KB

  grad_loss_partial<<<BLOCKS, THREADS, 0, stream>>>(img, objx, objy, kx, ky, mask, partial);
  grad_loss_finish<<<1, 256, 0, stream>>>(partial, out, BLOCKS);
}